// MeshPool_18863496364775
// MI455X (gfx1250) — compile-verified
//
#include <hip/hip_runtime.h>
#include <cstdint>

#define THREADS      256    // accum block (8 waves)
#define ETILE        1024   // edges per pipeline tile (4 per lane, 4 KB of ids)
#define CNT_THREADS  1024   // count block (32 waves)

typedef float v4f __attribute__((ext_vector_type(4)));
typedef int   v4i __attribute__((ext_vector_type(4)));

__device__ __forceinline__ void async_gid_tile(const int* __restrict__ gsrc,
                                               int* __restrict__ lds_dst) {
  // One CDNA5 async global->LDS b128 per lane (ASYNCcnt-tracked); each lane
  // later consumes exactly the 16 B it loaded, so only a per-wave
  // s_wait_asynccnt is needed -- no workgroup barrier in the pipeline.
  unsigned lds_off = (unsigned)(uintptr_t)lds_dst;
  unsigned long long ga = (unsigned long long)(uintptr_t)gsrc;
  asm volatile("global_load_async_to_lds_b128 %0, %1, off"
               :: "v"(lds_off), "v"(ga)
               : "memory");
}

// ---------------------------------------------------------------------------
// Per-batch group histogram in LDS, written densely to counts[b*T..].
// Fully overwrites the (poisoned) workspace region it owns -> no zero pass.
// ---------------------------------------------------------------------------
__global__ __launch_bounds__(CNT_THREADS) void count_kernel(const int* __restrict__ gid,
                                                            float* __restrict__ counts,
                                                            int E, int T) {
  extern __shared__ float s_cnt[];            // T floats
  const int b = blockIdx.x;
  for (int t = threadIdx.x; t < T; t += CNT_THREADS) s_cnt[t] = 0.0f;
  __syncthreads();
  const int* gb = gid + (size_t)b * (size_t)E;
  for (int i = threadIdx.x; i < E; i += CNT_THREADS) {
    unsafeAtomicAdd(&s_cnt[gb[i]], 1.0f);     // ds_add_f32
  }
  __syncthreads();
  float* cb = counts + (size_t)b * (size_t)T;
  for (int t = threadIdx.x; t < T; t += CNT_THREADS) cb[t] = s_cnt[t];
}

// ---------------------------------------------------------------------------
// One block per (b, c) output row. Segment-sum the 18000-edge feature row
// into an 18 KB LDS accumulator with ds_add_f32, then write the fused mean.
//  - ids: double-buffered async global->LDS pipeline (ASYNCcnt)
//  - fe: 16 B non-temporal streaming loads (keep 192 MB L2 for ids/counts/out)
//  - out: written exactly once, no global atomics anywhere in the hot path
// ---------------------------------------------------------------------------
__global__ __launch_bounds__(THREADS) void accum_kernel(const float* __restrict__ fe,
                                                        const int* __restrict__ gid,
                                                        const float* __restrict__ counts,
                                                        float* __restrict__ out,
                                                        int C, int E, int T) {
  extern __shared__ char smem[];
  float* s_sum = (float*)smem;                            // T floats
  const unsigned sum_bytes = ((unsigned)T * 4u + 15u) & ~15u;
  int* s_gid[2] = { (int*)(smem + sum_bytes),
                    (int*)(smem + sum_bytes) + ETILE };

  const int c = blockIdx.x;
  const int b = blockIdx.y;

  const int*   gb   = gid + (size_t)b * (size_t)E;
  const float* frow = fe  + ((size_t)b * C + c) * (size_t)E;

  for (int t = threadIdx.x; t < T; t += THREADS) s_sum[t] = 0.0f;

  const int ntiles = (E + ETILE - 1) / ETILE;
  const int lane4  = threadIdx.x * 4;

  // Prologue: start tile 0's id fetch while we hit the barrier for s_sum zero.
  if (lane4 < E) async_gid_tile(gb + lane4, s_gid[0] + lane4);
  __syncthreads();   // s_sum zeroing visible before any ds_add_f32

  for (int tile = 0; tile < ntiles; ++tile) {
    int* cur = s_gid[tile & 1];
    // Prefetch next tile, then wait for the current (older) one only.
    if (tile + 1 < ntiles) {
      int ee = (tile + 1) * ETILE + lane4;
      if (ee < E) async_gid_tile(gb + ee, s_gid[(tile + 1) & 1] + lane4);
      asm volatile("s_wait_asynccnt 0x1" ::: "memory");
    } else {
      asm volatile("s_wait_asynccnt 0x0" ::: "memory");
    }

    int e = tile * ETILE + lane4;
    if (e < E) {   // E % 4 == 0 -> all 4 elements of an active lane in range
      v4i g4 = *(const v4i*)(cur + lane4);                          // ds_load_b128
      v4f v  = __builtin_nontemporal_load((const v4f*)(frow + e));  // NT b128
      unsafeAtomicAdd(&s_sum[g4.x], v.x);                           // ds_add_f32
      unsafeAtomicAdd(&s_sum[g4.y], v.y);
      unsafeAtomicAdd(&s_sum[g4.z], v.z);
      unsafeAtomicAdd(&s_sum[g4.w], v.w);
    }
  }
  __syncthreads();   // all waves' LDS accumulation complete

  // Fused mean + single coalesced write of the output row.
  const float* cb  = counts + (size_t)b * (size_t)T;
  float*       orw = out + ((size_t)b * C + c) * (size_t)T;
  for (int t = threadIdx.x; t < T; t += THREADS) {
    orw[t] = s_sum[t] / fmaxf(cb[t], 1.0f);   // exact IEEE divide, like reference
  }
}

// ---------------------------------------------------------------------------
// Launch
// ---------------------------------------------------------------------------
extern "C" void kernel_launch(void* const* d_in, const int* in_sizes, int n_in,
                              void* d_out, int out_size, void* d_ws, size_t ws_size,
                              hipStream_t stream) {
  const float* fe  = (const float*)d_in[0];   // [B, C, E] float32
  const int*   gid = (const int*)d_in[1];     // [B, E]    int
  float* out    = (float*)d_out;              // [B, C, T] float32
  float* counts = (float*)d_ws;               // [B*T] floats of scratch

  const int B = 8;
  const long long BCE = (long long)in_sizes[0];   // B*C*E
  const int BE = in_sizes[1];                     // B*E
  const int C = (int)(BCE / (long long)BE);       // 256
  const int E = BE / B;                           // 18000
  const int T = out_size / (B * C);               // 4500

  // 1) per-batch group counts (LDS histogram, dense writeout)
  {
    size_t lds = (size_t)T * sizeof(float);                 // 18 KB
    count_kernel<<<dim3(B), CNT_THREADS, lds, stream>>>(gid, counts, E, T);
  }
  // 2) segment-sum + fused mean, one block per output row
  {
    size_t sum_bytes = (((size_t)T * 4 + 15) & ~(size_t)15);
    size_t lds = sum_bytes + 2 * ETILE * sizeof(int);       // 18 KB + 8 KB
    dim3 grid(C, B);
    accum_kernel<<<grid, THREADS, lds, stream>>>(fe, gid, counts, out, C, E, T);
  }
}